// RPEMultiHeadAttention_56203942035658
// MI455X (gfx1250) — compile-verified
//
#include <hip/hip_runtime.h>

#define DMODEL 256
#define HEADS  8
#define DHEAD  32
#define NSEQ   512
#define MSEQ   512

typedef float v2f __attribute__((ext_vector_type(2)));
typedef float v8f __attribute__((ext_vector_type(8)));

// ---------------------------------------------------------------------------
// Kernel 1: Y[n,o] = sum_i X[n,i]*W[o,i] + b[o]   (16 rows of n per block)
// ---------------------------------------------------------------------------
__global__ __launch_bounds__(256)
void proj16_kernel(const float* __restrict__ X, const float* __restrict__ W,
                   const float* __restrict__ bias, float* __restrict__ Y)
{
    __shared__ float Xs[16][DMODEL];
    const int n0 = blockIdx.x * 16;
    const int t  = threadIdx.x;                    // output column o
    for (int idx = t; idx < 16 * DMODEL; idx += 256)
        Xs[idx >> 8][idx & 255] = X[(size_t)(n0 + (idx >> 8)) * DMODEL + (idx & 255)];
    __syncthreads();

    float acc[16];
    const float bv = bias[t];
#pragma unroll
    for (int r = 0; r < 16; ++r) acc[r] = bv;
    const float* wrow = W + (size_t)t * DMODEL;
    for (int i = 0; i < DMODEL; ++i) {
        const float w = wrow[i];
#pragma unroll
        for (int r = 0; r < 16; ++r) acc[r] += Xs[r][i] * w;
    }
#pragma unroll
    for (int r = 0; r < 16; ++r) Y[(size_t)(n0 + r) * DMODEL + t] = acc[r];
}

// ---------------------------------------------------------------------------
// Kernel 2: build the per-n WMMA B-operand (K=512 x 8 heads) and bias term.
//   Ball[n][k][h], k<256 : qp[n,h,k]   = sum_c q[n,32h+c] * Wp[32h+c, k]
//   Ball[n][k][h], k>=256: qhat        = ((k-256)>>5 == h) ? q[n,k-256] : 0
//   qb[n][h] = sum_c q[n,32h+c] * bp[32h+c]
// ---------------------------------------------------------------------------
__global__ __launch_bounds__(256)
void build_b_kernel(const float* __restrict__ qf, const float* __restrict__ Wp,
                    const float* __restrict__ bp, float* __restrict__ Ball,
                    float* __restrict__ qb)
{
    __shared__ float qrow[DMODEL];
    const int n = blockIdx.x;
    const int t = threadIdx.x;                     // d index (0..255)
    qrow[t] = qf[(size_t)n * DMODEL + t];
    __syncthreads();

#pragma unroll
    for (int h = 0; h < HEADS; ++h) {
        float acc = 0.f;
#pragma unroll
        for (int c = 0; c < DHEAD; ++c)
            acc += qrow[h * DHEAD + c] * Wp[(size_t)(h * DHEAD + c) * DMODEL + t];
        Ball[((size_t)n * 512 + t) * 8 + h] = acc;
    }
    const int dh = t >> 5;                         // head owning column d
#pragma unroll
    for (int h = 0; h < HEADS; ++h)
        Ball[((size_t)n * 512 + 256 + t) * 8 + h] = (h == dh) ? qrow[t] : 0.f;

    if (t < HEADS) {
        float acc = 0.f;
#pragma unroll
        for (int c = 0; c < DHEAD; ++c)
            acc += qrow[t * DHEAD + c] * bp[t * DHEAD + c];
        qb[(size_t)n * HEADS + t] = acc;
    }
}

// ---------------------------------------------------------------------------
// Kernel 3: per-n fused scores (WMMA f32 16x16x4) + softmax + attn@v.
//   D[m,h] = sum_{k<256} e[n,m,k]*qp[n,h,k] + sum_{k<256} kf[m,k]*qhat[k,h]
// One block per n; 8 waves; each wave owns 4 m-tiles of 16.
// ---------------------------------------------------------------------------
#define BS_STRIDE 516   // pad so Bs[col][k] b64 reads are bank-conflict-free

__global__ __launch_bounds__(256)
void rpe_attn_kernel(const float* __restrict__ e,   // embed_qk (N,M,D)
                     const float* __restrict__ kf,  // (M,D)
                     const float* __restrict__ vf,  // (M,D)
                     const float* __restrict__ Ball,// (N,512,8)
                     const float* __restrict__ qb,  // (N,8)
                     float* __restrict__ out_hidden,// (N,256)
                     float* __restrict__ out_attn)  // (H,N,M)
{
    __shared__ float Bs[16 * BS_STRIDE];           // B_cat, col-major, 16 cols
    __shared__ float attn_s[HEADS * MSEQ];
    __shared__ float red_max[16][8];
    __shared__ float red_sum[16][8];

    const int n    = blockIdx.x;
    const int tid  = threadIdx.x;
    const int wave = tid >> 5;
    const int lane = tid & 31;
    const int col  = lane & 15;                    // WMMA N index (head)
    const int hi   = lane >> 4;                    // lane half
    const int kb   = hi * 2;                       // K pair base {0,1}/{2,3}

    // Load + zero-pad B_cat into LDS: Bs[col*516 + k]
    const float* Bn = Ball + (size_t)n * 512 * 8;
    for (int idx = tid; idx < 512 * 16; idx += 256) {
        const int k = idx >> 4, c = idx & 15;
        Bs[c * BS_STRIDE + k] = (c < 8) ? Bn[k * 8 + c] : 0.f;
    }
    __syncthreads();

    // A-operand row pointers (A row index = col per ISA 16x4 layout)
    const float* ae[4];
    const float* ak[4];
#pragma unroll
    for (int t = 0; t < 4; ++t) {
        const int m0 = wave * 64 + t * 16;
        ae[t] = e  + ((size_t)n * MSEQ + m0 + col) * DMODEL + kb;
        ak[t] = kf + (size_t)(m0 + col) * DMODEL + kb;
    }

    v8f acc[4];
#pragma unroll
    for (int t = 0; t < 4; ++t) acc[t] = (v8f){0,0,0,0,0,0,0,0};

    // K = 0..255 : embed_qk against qp
    for (int k0 = 0; k0 < 256; k0 += 4) {
        const v2f b = *(const v2f*)&Bs[col * BS_STRIDE + k0 + kb];
#pragma unroll
        for (int t = 0; t < 4; ++t) {
            const v2f a = *(const v2f*)(ae[t] + k0);
            acc[t] = __builtin_amdgcn_wmma_f32_16x16x4_f32(
                false, a, false, b, (short)0, acc[t], false, false);
        }
    }
    // K = 256..511 : k_full against head-masked q  (adds scores_e)
    for (int k0 = 0; k0 < 256; k0 += 4) {
        const v2f b = *(const v2f*)&Bs[col * BS_STRIDE + 256 + k0 + kb];
#pragma unroll
        for (int t = 0; t < 4; ++t) {
            const v2f a = *(const v2f*)(ak[t] + k0);
            acc[t] = __builtin_amdgcn_wmma_f32_16x16x4_f32(
                false, a, false, b, (short)0, acc[t], false, false);
        }
    }

    // scores = (acc + qb) / sqrt(32); D layout: lane=(col,hi), vgpr j -> m = m0+j+8*hi
    const float qbv   = (col < HEADS) ? qb[(size_t)n * HEADS + col] : 0.f;
    const float scale = 0.17677669529663687f;      // 1/sqrt(32)

    float mloc = -3.4e38f;
#pragma unroll
    for (int t = 0; t < 4; ++t)
#pragma unroll
        for (int j = 0; j < 8; ++j) {
            const float s = (acc[t][j] + qbv) * scale;
            acc[t][j] = s;
            mloc = fmaxf(mloc, s);
        }
    mloc = fmaxf(mloc, __shfl_xor(mloc, 16, 32));
    if (lane < 16) red_max[col][wave] = mloc;
    __syncthreads();
    float M = -3.4e38f;
#pragma unroll
    for (int w = 0; w < 8; ++w) M = fmaxf(M, red_max[col][w]);

    float sloc = 0.f;
#pragma unroll
    for (int t = 0; t < 4; ++t)
#pragma unroll
        for (int j = 0; j < 8; ++j) {
            const float ex = __expf(acc[t][j] - M);
            acc[t][j] = ex;
            sloc += ex;
        }
    sloc += __shfl_xor(sloc, 16, 32);
    if (lane < 16) red_sum[col][wave] = sloc;
    __syncthreads();
    float S = 0.f;
#pragma unroll
    for (int w = 0; w < 8; ++w) S += red_sum[col][w];
    const float inv = 1.0f / S;

    if (col < HEADS) {
#pragma unroll
        for (int t = 0; t < 4; ++t) {
            const int m0 = wave * 64 + t * 16;
#pragma unroll
            for (int j = 0; j < 8; ++j) {
                const int m = m0 + j + 8 * hi;
                const float a = acc[t][j] * inv;
                attn_s[col * MSEQ + m] = a;
                out_attn[((size_t)col * NSEQ + n) * MSEQ + m] = a;
            }
        }
    }
    __syncthreads();

    // hidden[n, h*32+c] = sum_m attn[h,m] * v[m, h*32+c]
    const int h = tid >> 5;
    float hv = 0.f;
    const float* vcol = vf + tid;
    for (int m = 0; m < MSEQ; ++m)
        hv += attn_s[h * MSEQ + m] * vcol[(size_t)m * DMODEL];
    out_hidden[(size_t)n * DMODEL + tid] = hv;
}

// ---------------------------------------------------------------------------
extern "C" void kernel_launch(void* const* d_in, const int* in_sizes, int n_in,
                              void* d_out, int out_size, void* d_ws, size_t ws_size,
                              hipStream_t stream) {
    (void)in_sizes; (void)n_in; (void)out_size; (void)ws_size;
    const float* input_q = (const float*)d_in[0];
    const float* input_k = (const float*)d_in[1];
    const float* input_v = (const float*)d_in[2];
    const float* embed   = (const float*)d_in[3];
    const float* Wq = (const float*)d_in[4];  const float* bq = (const float*)d_in[5];
    const float* Wk = (const float*)d_in[6];  const float* bk = (const float*)d_in[7];
    const float* Wv = (const float*)d_in[8];  const float* bv = (const float*)d_in[9];
    const float* Wp = (const float*)d_in[10]; const float* bp = (const float*)d_in[11];

    float* ws   = (float*)d_ws;
    float* qf   = ws;                    // 512*256
    float* kf   = ws + 131072;           // 512*256
    float* vf   = ws + 262144;           // 512*256
    float* qb   = ws + 393216;           // 512*8
    float* Ball = ws + 397312;           // 512*512*8  (~8 MB)

    float* out_hidden = (float*)d_out;               // (N, 256)
    float* out_attn   = (float*)d_out + NSEQ * DMODEL; // (H, N, M)

    proj16_kernel<<<NSEQ / 16, 256, 0, stream>>>(input_q, Wq, bq, qf);
    proj16_kernel<<<NSEQ / 16, 256, 0, stream>>>(input_k, Wk, bk, kf);
    proj16_kernel<<<NSEQ / 16, 256, 0, stream>>>(input_v, Wv, bv, vf);
    build_b_kernel<<<NSEQ, 256, 0, stream>>>(qf, Wp, bp, Ball, qb);
    rpe_attn_kernel<<<NSEQ, 256, 0, stream>>>(embed, kf, vf, Ball, qb,
                                              out_hidden, out_attn);
}